// MolOptDecoder_27900107555247
// MI455X (gfx1250) — compile-verified
//
#include <hip/hip_runtime.h>

// CDNA5 / gfx1250, wave32. f32 WMMA 16x16x4 throughout (reference is f32).

typedef __attribute__((ext_vector_type(2))) float v2f;
typedef __attribute__((ext_vector_type(8))) float v8f;

#define HD 128          // hidden dim H == P
#define LA 48           // atoms per molecule
#define NSYM 12
#define NCHG 5
#define NBND 5

__device__ __forceinline__ v8f wmma_k4(v2f a, v2f b, v8f c) {
  // D = A(16x4, f32) * B(4x16, f32) + C(16x16, f32)
  return __builtin_amdgcn_wmma_f32_16x16x4_f32(false, a, false, b, (short)0, c,
                                               false, false);
}

// ---------------------------------------------------------------------------
// Weight pre-transposition into WMMA B-operand layout.
// Layer-1 (128x128): w1t[((kc*8+nt)*32+lane)*2 + {0,1}] = W[(4kc+2half)+{0,1}][nt*16+lrow]
// Layer-2 (128xOD, OD<=16, zero-padded): w2t[(kc*32+lane)*2 + {0,1}]
// ---------------------------------------------------------------------------
__global__ __launch_bounds__(256) void prep_w1_kernel(
    const float* __restrict__ W1s, const float* __restrict__ W1c,
    const float* __restrict__ W1b, float* __restrict__ w1t) {
  const int kc = blockIdx.x;   // 0..31
  const int mat = blockIdx.y;  // 0..3
  const float* W = (mat == 0) ? W1s
                 : (mat == 1) ? W1c
                 : (mat == 2) ? W1b
                              : (W1b + HD * HD);
  float* out = w1t + mat * (HD * HD);
  const int nt = threadIdx.x >> 5;
  const int lane = threadIdx.x & 31;
  const int half = lane >> 4, lrow = lane & 15;
  const int r0 = 4 * kc + 2 * half;
  v2f w;
  w.x = W[r0 * HD + nt * 16 + lrow];
  w.y = W[(r0 + 1) * HD + nt * 16 + lrow];
  *(v2f*)(out + ((kc * 8 + nt) * 32 + lane) * 2) = w;
}

__global__ __launch_bounds__(32) void prep_w2_kernel(
    const float* __restrict__ W2s, const float* __restrict__ W2c,
    const float* __restrict__ W2b, float* __restrict__ w2t) {
  const int mat = blockIdx.x;  // 0: W2s(12), 1: W2c(5), 2: W2b(5)
  const float* W = (mat == 0) ? W2s : (mat == 1) ? W2c : W2b;
  const int outdim = (mat == 0) ? NSYM : NCHG;
  float* out = w2t + mat * 2048;
  const int lane = threadIdx.x;
  const int half = lane >> 4, lrow = lane & 15;
  const int cw = lrow < outdim ? lrow : (outdim - 1);  // in-bounds clamp
  for (int kc = 0; kc < 32; ++kc) {
    const int r0 = 4 * kc + 2 * half;
    v2f w;
    w.x = lrow < outdim ? W[r0 * outdim + cw] : 0.f;
    w.y = lrow < outdim ? W[(r0 + 1) * outdim + cw] : 0.f;
    *(v2f*)(out + (kc * 32 + lane) * 2) = w;
  }
}

// ---- layer-1 GEMM for one 16-atom tile: 16x128 accumulators -----------------
__device__ __forceinline__ void gemm16x128(const float* __restrict__ xrow,
                                           const float* __restrict__ W1t,
                                           int lane, int koff, v8f acc[8]) {
#pragma unroll
  for (int nt = 0; nt < 8; ++nt) acc[nt] = (v8f){};
  for (int kc = 0; kc < 32; ++kc) {
    const v2f a = *(const v2f*)(xrow + 4 * kc + koff);
#pragma unroll
    for (int nt = 0; nt < 8; ++nt) {
      const v2f bw = *(const v2f*)(W1t + ((kc * 8 + nt) * 32 + lane) * 2);
      acc[nt] = wmma_k4(a, bw, acc[nt]);
    }
  }
}

// ---- full 2-layer head (symbols / charges) ---------------------------------
__device__ __forceinline__ void head_mlp(const float* __restrict__ xrow,
                                         const float* __restrict__ W1t,
                                         const float* __restrict__ b1,
                                         const float* __restrict__ W2t,
                                         const float* __restrict__ b2,
                                         int outdim, float* __restrict__ myhid,
                                         float* __restrict__ out, int atom0,
                                         int lane, int lrow, int half,
                                         int koff) {
  v8f acc[8];
  gemm16x128(xrow, W1t, lane, koff, acc);
  // relu(+bias) -> LDS row-major [16][128]
#pragma unroll
  for (int nt = 0; nt < 8; ++nt) {
    const int col = nt * 16 + lrow;
    const float bias = b1[col];
#pragma unroll
    for (int r = 0; r < 8; ++r) {
      float hv = acc[nt][r] + bias;
      myhid[(r + 8 * half) * HD + col] = hv > 0.f ? hv : 0.f;
    }
  }
  asm volatile("" ::: "memory");  // same-wave DS ordering is in-order by HW
  // layer 2: hid(16x128) @ W2(128xoutdim), columns >= outdim are zero-padded
  v8f acc2 = {};
  for (int kc = 0; kc < 32; ++kc) {
    const v2f a2 = *(const v2f*)(myhid + lrow * HD + 4 * kc + koff);
    const v2f bw = *(const v2f*)(W2t + (kc * 32 + lane) * 2);
    acc2 = wmma_k4(a2, bw, acc2);
  }
  if (lrow < outdim) {
    const float bias = b2[lrow];
#pragma unroll
    for (int r = 0; r < 8; ++r)
      out[(atom0 + r + 8 * half) * outdim + lrow] = acc2[r] + bias;
  }
}

// ---- linear head (u / v precompute); optional fused bias -------------------
__device__ __forceinline__ void head_lin(const float* __restrict__ xrow,
                                         const float* __restrict__ W1t,
                                         const float* __restrict__ b1,
                                         float* __restrict__ out, int atom0,
                                         int lane, int lrow, int half,
                                         int koff) {
  v8f acc[8];
  gemm16x128(xrow, W1t, lane, koff, acc);
#pragma unroll
  for (int nt = 0; nt < 8; ++nt) {
    const int col = nt * 16 + lrow;
    const float bias = b1 ? b1[col] : 0.f;
#pragma unroll
    for (int r = 0; r < 8; ++r)
      out[(atom0 + r + 8 * half) * HD + col] = acc[nt][r] + bias;
  }
}

__global__ __launch_bounds__(256) void atom_heads_kernel(
    const float* __restrict__ x, const float* __restrict__ w1t,
    const float* __restrict__ w2t, const float* __restrict__ b1s,
    const float* __restrict__ b2s, const float* __restrict__ b1c,
    const float* __restrict__ b2c, const float* __restrict__ b1b,
    float* __restrict__ out_sym, float* __restrict__ out_chg,
    float* __restrict__ u, float* __restrict__ v) {
  __shared__ float hid[8][16 * HD];  // 8 KB per wave, 64 KB per block
  const int lane = threadIdx.x & 31;
  const int wave = threadIdx.x >> 5;
  const int half = lane >> 4;
  const int lrow = lane & 15;
  const int koff = half * 2;
  const int atom0 = (blockIdx.x * 8 + wave) * 16;
  const float* xrow = x + (atom0 + lrow) * HD;
  float* myhid = hid[wave];

  head_mlp(xrow, w1t, b1s, w2t, b2s, NSYM, myhid, out_sym, atom0, lane, lrow,
           half, koff);
  head_mlp(xrow, w1t + HD * HD, b1c, w2t + 2048, b2c, NCHG, myhid, out_chg,
           atom0, lane, lrow, half, koff);
  // u = x @ W1b_top + b1b (bias folded here); v = x @ W1b_bot
  head_lin(xrow, w1t + 2 * HD * HD, b1b, u, atom0, lane, lrow, half, koff);
  head_lin(xrow, w1t + 3 * HD * HD, nullptr, v, atom0, lane, lrow, half, koff);
}

// One wave per (molecule b, row-atom l): all 48 m's (3 tiles of 16).
// bonds[b,l,m,:] = relu(u'[b,l] + v[b,m]) @ W2b + b2b   (u' has b1b folded)
__global__ __launch_bounds__(256) void bond_pairs_kernel(
    const float* __restrict__ u, const float* __restrict__ v,
    const float* __restrict__ w2bt, const float* __restrict__ b2b,
    float* __restrict__ out_bonds) {
  const int lane = threadIdx.x & 31;
  const int wave = threadIdx.x >> 5;
  const int half = lane >> 4;
  const int lrow = lane & 15;
  const int koff = half * 2;
  const int tid = blockIdx.x * 8 + wave;  // 0 .. 12287  == b*48 + l
  const int b = tid / LA;
  const int l = tid % LA;

  const float* urow = u + (b * LA + l) * HD;       // broadcast row
  const float* vbase = v + (b * LA + lrow) * HD;   // per-pair rows, tile 0

  v8f acc0 = {}, acc1 = {}, acc2 = {};
  for (int kc = 0; kc < 32; ++kc) {
    const int r0 = 4 * kc + koff;
    const v2f uv = *(const v2f*)(urow + r0);
    const v2f bw = *(const v2f*)(w2bt + (kc * 32 + lane) * 2);
    const v2f v0 = *(const v2f*)(vbase + r0);
    const v2f v1 = *(const v2f*)(vbase + 16 * HD + r0);
    const v2f v2 = *(const v2f*)(vbase + 32 * HD + r0);
    v2f a0, a1, a2;
    a0.x = fmaxf(uv.x + v0.x, 0.f); a0.y = fmaxf(uv.y + v0.y, 0.f);
    a1.x = fmaxf(uv.x + v1.x, 0.f); a1.y = fmaxf(uv.y + v1.y, 0.f);
    a2.x = fmaxf(uv.x + v2.x, 0.f); a2.y = fmaxf(uv.y + v2.y, 0.f);
    acc0 = wmma_k4(a0, bw, acc0);
    acc1 = wmma_k4(a1, bw, acc1);
    acc2 = wmma_k4(a2, bw, acc2);
  }
  if (lrow < NBND) {
    const float bias = b2b[lrow];
    const int base = (b * LA + l) * LA + 8 * half;
#pragma unroll
    for (int r = 0; r < 8; ++r) {
      out_bonds[(base + r) * NBND + lrow] = acc0[r] + bias;
      out_bonds[(base + 16 + r) * NBND + lrow] = acc1[r] + bias;
      out_bonds[(base + 32 + r) * NBND + lrow] = acc2[r] + bias;
    }
  }
}

extern "C" void kernel_launch(void* const* d_in, const int* in_sizes, int n_in,
                              void* d_out, int out_size, void* d_ws,
                              size_t ws_size, hipStream_t stream) {
  (void)in_sizes; (void)n_in; (void)out_size; (void)ws_size;
  const float* x = (const float*)d_in[0];
  // d_in[1] = scope (int64), uniform -> unused
  const float* W1s = (const float*)d_in[2];
  const float* b1s = (const float*)d_in[3];
  const float* W2s = (const float*)d_in[4];
  const float* b2s = (const float*)d_in[5];
  const float* W1c = (const float*)d_in[6];
  const float* b1c = (const float*)d_in[7];
  const float* W2c = (const float*)d_in[8];
  const float* b2c = (const float*)d_in[9];
  const float* W1b = (const float*)d_in[10];
  const float* b1b = (const float*)d_in[11];
  const float* W2b = (const float*)d_in[12];
  const float* b2b = (const float*)d_in[13];

  const int N = 256 * LA;  // 12288 atoms
  float* out_sym = (float*)d_out;        // N * 12
  float* out_chg = out_sym + N * NSYM;   // N * 5
  float* out_bnd = out_chg + N * NCHG;   // 256*48*48 * 5

  float* u = (float*)d_ws;               // N * 128
  float* v = u + N * HD;                 // N * 128
  float* w1t = v + N * HD;               // 4 * 128*128 (B-layout weights)
  float* w2t = w1t + 4 * HD * HD;        // 3 * 2048    (padded B-layout)

  prep_w1_kernel<<<dim3(32, 4), 256, 0, stream>>>(W1s, W1c, W1b, w1t);
  prep_w2_kernel<<<3, 32, 0, stream>>>(W2s, W2c, W2b, w2t);
  atom_heads_kernel<<<96, 256, 0, stream>>>(x, w1t, w2t, b1s, b2s, b1c, b2c,
                                            b1b, out_sym, out_chg, u, v);
  bond_pairs_kernel<<<1536, 256, 0, stream>>>(u, v, w2t + 4096, b2b, out_bnd);
}